// RNN_68470368633224
// MI455X (gfx1250) — compile-verified
//
#include <hip/hip_runtime.h>
#include <stdint.h>

typedef __attribute__((ext_vector_type(2))) float v2f;
typedef __attribute__((ext_vector_type(8))) float v8f;

#define TILE_M 64
#define TILE_N 128
#define TILE_K 32
#define LDSP   36   // 32 + 4 pad: rows stay 16B-aligned (144B); 36*r mod 64 conflict-free

#define ASYNC_WAIT_6() asm volatile("s_wait_asynccnt 0x6" ::: "memory")
#define ASYNC_WAIT_0() asm volatile("s_wait_asynccnt 0x0" ::: "memory")

// Flexible fp32 WMMA GEMM:
//   out[m][n] = act( sum_k X1[row(m)][k]*W1[n][k]  (+ sum_k X2[m][k]*W2[n][k])
//                    + bias[n] + add[m*N+n] )
// X row-major (ld=K), W row-major (N x K)  => computes X @ W^T.
// xidx gathers X1 rows (fused embedding lookup).  streaming=1 -> NT stores.
// Interior blocks: double-buffered GLOBAL_LOAD_ASYNC_TO_LDS_B128 staging.
// Each of 8 waves owns a 32x32 output tile: 4 x ds_load_b64 -> 4 x v_wmma per k-step.
__global__ __launch_bounds__(256) void gemm_wmma_f32(
    const float* __restrict__ X1, const int* __restrict__ xidx, int K1,
    const float* __restrict__ W1,
    const float* __restrict__ X2, int K2, const float* __restrict__ W2,
    const float* __restrict__ bias, const float* __restrict__ add,
    float* __restrict__ out, int M, int N, int apply_tanh, int streaming)
{
    __shared__ float lA[2][TILE_M][LDSP];
    __shared__ float lB[2][TILE_N][LDSP];

    const int tid  = threadIdx.x;
    const int lane = tid & 31;
    const int wave = tid >> 5;
    const int wm   = wave >> 2;      // 0..1 : 32-row strip
    const int wn   = wave & 3;       // 0..3 : 32-col strip
    const int m0   = blockIdx.x * TILE_M;
    const int n0   = blockIdx.y * TILE_N;

    v8f acc00 = {}, acc01 = {}, acc10 = {}, acc11 = {};

    // cooperative staging coordinates
    const int arow = tid >> 2;          // 0..63   A: 2 x b128 per thread
    const int acol = (tid & 3) << 2;    // 0,4,8,12  (+16 for 2nd chunk)
    const int brow = tid >> 1;          // 0..127  B: 4 x b128 per thread
    const int bcol = (tid & 1) << 4;    // 0,16

    const int mrow = (wm << 5) + (lane & 15);   // A fragment: lane&15 = M
    const int nrow = (wn << 5) + (lane & 15);   // B fragment: lane&15 = N
    const bool full_n = (n0 + TILE_N <= N);

    auto compute_tile = [&](int buf) {
        #pragma unroll
        for (int ks = 0; ks < TILE_K / 4; ++ks) {
            const int kb = (ks << 2) + ((lane >> 4) << 1); // lanes 16-31: K+2,K+3
            v2f a0 = *(const v2f*)&lA[buf][mrow     ][kb]; // single ds_load_b64 each
            v2f a1 = *(const v2f*)&lA[buf][mrow + 16][kb];
            v2f b0 = *(const v2f*)&lB[buf][nrow     ][kb];
            v2f b1 = *(const v2f*)&lB[buf][nrow + 16][kb];
            acc00 = __builtin_amdgcn_wmma_f32_16x16x4_f32(false, a0, false, b0, (short)0, acc00, false, false);
            acc01 = __builtin_amdgcn_wmma_f32_16x16x4_f32(false, a0, false, b1, (short)0, acc01, false, false);
            acc10 = __builtin_amdgcn_wmma_f32_16x16x4_f32(false, a1, false, b0, (short)0, acc10, false, false);
            acc11 = __builtin_amdgcn_wmma_f32_16x16x4_f32(false, a1, false, b1, (short)0, acc11, false, false);
        }
    };

    for (int term = 0; term < 2; ++term) {
        const float* Xp = term ? X2 : X1;
        const float* Wp = term ? W2 : W1;
        const int    K  = term ? K2 : K1;
        if (Xp == nullptr || K == 0) continue;

        long xrow = (long)(m0 + arow);                     // hoisted row gather
        if (term == 0 && xidx) xrow = (long)xidx[m0 + arow];
        const float* aBase = Xp + xrow * (long)K + acol;
        const float* bBase = Wp + (long)(n0 + brow) * (long)K + bcol;
        const int nk = K / TILE_K;

        if (full_n) {
            // ---- async double-buffered pipeline: 6 async ops per tile ----
            auto issue_tile = [&](int kt, int buf) {
                const float*   gA = aBase + kt * TILE_K;
                const float*   gB = bBase + kt * TILE_K;
                const uint32_t la = (uint32_t)(uintptr_t)&lA[buf][arow][acol];
                const uint32_t lb = (uint32_t)(uintptr_t)&lB[buf][brow][bcol];
                // offset: is added to BOTH the LDS and global address (ISA 8.4.4)
                asm volatile("global_load_async_to_lds_b128 %0, %1, off"           :: "v"(la), "v"(gA) : "memory");
                asm volatile("global_load_async_to_lds_b128 %0, %1, off offset:64" :: "v"(la), "v"(gA) : "memory");
                asm volatile("global_load_async_to_lds_b128 %0, %1, off"           :: "v"(lb), "v"(gB) : "memory");
                asm volatile("global_load_async_to_lds_b128 %0, %1, off offset:16" :: "v"(lb), "v"(gB) : "memory");
                asm volatile("global_load_async_to_lds_b128 %0, %1, off offset:32" :: "v"(lb), "v"(gB) : "memory");
                asm volatile("global_load_async_to_lds_b128 %0, %1, off offset:48" :: "v"(lb), "v"(gB) : "memory");
            };

            issue_tile(0, 0);
            for (int i = 0; i < nk; i += 2) {             // 2x unroll: literal buf index
                if (i + 1 < nk) { issue_tile(i + 1, 1); ASYNC_WAIT_6(); }
                else            { ASYNC_WAIT_0(); }
                __syncthreads();
                compute_tile(0);
                __syncthreads();
                if (i + 1 >= nk) break;
                if (i + 2 < nk) { issue_tile(i + 2, 0); ASYNC_WAIT_6(); }
                else            { ASYNC_WAIT_0(); }
                __syncthreads();
                compute_tile(1);
                __syncthreads();
            }
        } else {
            // ---- guarded synchronous path (N-tail blocks only) ----
            for (int kt = 0; kt < nk; ++kt) {
                const int k0 = kt * TILE_K;
                {
                    const float4 va0 = *(const float4*)(aBase + k0);
                    const float4 va1 = *(const float4*)(aBase + k0 + 16);
                    lA[0][arow][acol+0]  = va0.x; lA[0][arow][acol+1]  = va0.y;
                    lA[0][arow][acol+2]  = va0.z; lA[0][arow][acol+3]  = va0.w;
                    lA[0][arow][acol+16] = va1.x; lA[0][arow][acol+17] = va1.y;
                    lA[0][arow][acol+18] = va1.z; lA[0][arow][acol+19] = va1.w;
                }
                {
                    float4 vb[4];
                    #pragma unroll
                    for (int q = 0; q < 4; ++q) vb[q] = make_float4(0.f, 0.f, 0.f, 0.f);
                    if (n0 + brow < N) {
                        #pragma unroll
                        for (int q = 0; q < 4; ++q)
                            vb[q] = *(const float4*)(bBase + k0 + 4 * q);
                    }
                    #pragma unroll
                    for (int q = 0; q < 4; ++q) {
                        lB[0][brow][bcol + 4*q + 0] = vb[q].x;
                        lB[0][brow][bcol + 4*q + 1] = vb[q].y;
                        lB[0][brow][bcol + 4*q + 2] = vb[q].z;
                        lB[0][brow][bcol + 4*q + 3] = vb[q].w;
                    }
                }
                __syncthreads();
                compute_tile(0);
                __syncthreads();
            }
        }
    }

    // ---- epilogue: C/D layout = VGPR r -> rows r (lanes 0-15) / r+8 (lanes 16-31) ----
    const int nb = n0 + (wn << 5) + (lane & 15);
    const int mb = m0 + (wm << 5) + ((lane >> 4) << 3);
    auto emit = [&](const v8f& acc, int mofs, int nofs) {
        const int n = nb + nofs;
        if (n >= N) return;
        const float bv = bias ? bias[n] : 0.f;
        #pragma unroll
        for (int r = 0; r < 8; ++r) {
            const long m = mb + mofs + r;
            float v = acc[r] + bv;
            if (add)        v += add[m * (long)N + n];
            if (apply_tanh) v  = tanhf(v);
            if (streaming)  __builtin_nontemporal_store(v, out + m * (long)N + n);
            else            out[m * (long)N + n] = v;
        }
    };
    emit(acc00,  0,  0);
    emit(acc01,  0, 16);
    emit(acc10, 16,  0);
    emit(acc11, 16, 16);
}

// Pack final hidden state: dst[0:n] = h0_last, dst[n:2n] = h1_last.
__global__ void pack_hidden(const float* __restrict__ h0, const float* __restrict__ h1,
                            float* __restrict__ dst, int n)
{
    const int i = blockIdx.x * blockDim.x + threadIdx.x;
    if (i < n)            dst[i] = h0[i];
    else if (i < 2 * n)   dst[i] = h1[i - n];
}

extern "C" void kernel_launch(void* const* d_in, const int* in_sizes, int n_in,
                              void* d_out, int out_size, void* d_ws, size_t ws_size,
                              hipStream_t stream)
{
    (void)in_sizes; (void)n_in; (void)out_size; (void)ws_size;

    const int*   inputs = (const int*)  d_in[0];   // (T,B) int32
    const float* hidden = (const float*)d_in[1];   // (2,B,H)
    const float* embW   = (const float*)d_in[2];   // (V,E)
    const float* Wx0    = (const float*)d_in[3];   // (H,E)
    const float* Wx1    = (const float*)d_in[4];   // (H,H)
    const float* Wh0    = (const float*)d_in[5];   // (H,H)
    const float* b0     = (const float*)d_in[6];   // (H)
    const float* Wh1    = (const float*)d_in[7];   // (H,H)
    const float* b1     = (const float*)d_in[8];   // (H)
    const float* Wy     = (const float*)d_in[9];   // (V,H)
    const float* by     = (const float*)d_in[10];  // (V)
    float* out = (float*)d_out;                    // logits (T,B,V) ++ hidden (2,B,H)

    const int T = 64, B = 128, E = 1024, H = 1024, V = 10000;
    const long BH = (long)B * H;

    float* A0  = (float*)d_ws;          // (T*B, H): x@Wx0^T + b0, all timesteps
    float* H1  = A0 + (long)T * BH;     // (T*B, H): h1 for all timesteps
    float* h0p = H1 + (long)T * BH;     // 2 ping-pong h0 buffers (B,H)

    const dim3 blk(256);

    // 1) A0 = emb_W[inputs] @ Wx0^T + b0   (embedding gather fused via xidx)
    gemm_wmma_f32<<<dim3((T * B) / TILE_M, H / TILE_N), blk, 0, stream>>>(
        embW, inputs, E, Wx0, nullptr, 0, nullptr, b0, nullptr,
        A0, T * B, H, /*tanh=*/0, /*stream=*/0);

    // 2) serial recurrence, 2 launches per step
    for (int t = 0; t < T; ++t) {
        const float* h0prev = (t == 0) ? hidden : h0p + ((t - 1) & 1) * BH;
        float*       h0cur  = h0p + (t & 1) * BH;
        // h0 = tanh(h0prev @ Wh0^T + A0[t])
        gemm_wmma_f32<<<dim3(B / TILE_M, H / TILE_N), blk, 0, stream>>>(
            h0prev, nullptr, H, Wh0, nullptr, 0, nullptr,
            nullptr, A0 + (long)t * BH, h0cur, B, H, /*tanh=*/1, 0);

        const float* h1prev = (t == 0) ? hidden + BH : H1 + (long)(t - 1) * BH;
        // h1 = tanh(h0cur @ Wx1^T + h1prev @ Wh1^T + b1)   (dual-term GEMM)
        gemm_wmma_f32<<<dim3(B / TILE_M, H / TILE_N), blk, 0, stream>>>(
            h0cur, nullptr, H, Wx1, h1prev, H, Wh1,
            b1, nullptr, H1 + (long)t * BH, B, H, /*tanh=*/1, 0);
    }

    // 3) all logits in one big GEMM: (T*B,H) @ (V,H)^T + by, streamed to HBM (NT stores)
    gemm_wmma_f32<<<dim3((T * B) / TILE_M, (V + TILE_N - 1) / TILE_N), blk, 0, stream>>>(
        H1, nullptr, H, Wy, nullptr, 0, nullptr, by, nullptr,
        out, T * B, V, /*tanh=*/0, /*stream=*/1);

    // 4) final hidden = stack([h0_last, h1_last]) appended after logits
    const float* h0last = h0p + ((T - 1) & 1) * BH;
    const float* h1last = H1 + (long)(T - 1) * BH;
    pack_hidden<<<dim3((2 * (int)BH + 255) / 256), blk, 0, stream>>>(
        h0last, h1last, out + (long)T * B * V, (int)BH);
}